// Model_86535001080606
// MI455X (gfx1250) — compile-verified
//
#include <hip/hip_runtime.h>

// ---------------------------------------------------------------------------
// 2-layer LSTM forecaster on gfx1250 (MI455X)
// fp32 WMMA (V_WMMA_F32_16X16X4_F32) + async global->LDS staging + reg pipelining
// ---------------------------------------------------------------------------

typedef __attribute__((ext_vector_type(2))) float v2f;
typedef __attribute__((ext_vector_type(8))) float v8f;
typedef __attribute__((ext_vector_type(4))) int v4i;

#define B_    128
#define T_    1024
#define F_    64
#define H_    256
#define PRED_ 96
#define G4H_  1024   // 4*H

#if __has_builtin(__builtin_amdgcn_global_load_async_to_lds_b128)
#define HAVE_ASYNC_LDS 1
typedef __attribute__((address_space(1))) v4i g1_v4i;  // global int4*
typedef __attribute__((address_space(3))) v4i l3_v4i;  // LDS int4*
#endif

__device__ __forceinline__ void async_wait_all() {
#if defined(HAVE_ASYNC_LDS)
#if __has_builtin(__builtin_amdgcn_s_wait_asynccnt)
  __builtin_amdgcn_s_wait_asynccnt(0);
#else
  asm volatile("s_wait_asynccnt 0x0" ::: "memory");
#endif
#endif
}

// copy one 16-byte chunk global -> LDS (async path if available)
__device__ __forceinline__ void stage16(const float* g, float* l) {
#if defined(HAVE_ASYNC_LDS)
  __builtin_amdgcn_global_load_async_to_lds_b128((g1_v4i*)g, (l3_v4i*)l, 0, 0);
#else
  *(float4*)l = *(const float4*)g;
#endif
}

__device__ __forceinline__ v8f v8zero() {
  v8f v;
#pragma unroll
  for (int i = 0; i < 8; ++i) v[i] = 0.f;
  return v;
}

__device__ __forceinline__ float sigf(float x) { return 1.0f / (1.0f + __expf(-x)); }

// fp32 WMMA: D(16x16,f32) = A(16x4,f32) * B(4x16,f32) + C
// A frag: lane m=lane%16, half=lane/16 -> a[j] = A[m][2*half+j]
// B frag: lane n=lane%16              -> b[j] = B[2*half+j][n]
// C/D:    elem e -> row = e + 8*half, col = lane%16
__device__ __forceinline__ v8f wmma4(v2f a, v2f b, v8f c) {
  return __builtin_amdgcn_wmma_f32_16x16x4_f32(false, a, false, b, (short)0, c,
                                               false, false);
}

// ---------------------------------------------------------------------------
// prep: Wc[4H,F] = W_ih0[4H,H] @ W_hid[H,F];  bc = W_ih0@b_hid + b_ih0 + b_hh0
// ---------------------------------------------------------------------------
__global__ void __launch_bounds__(64) prep_kernel(
    const float* __restrict__ W_ih0, const float* __restrict__ W_hid,
    const float* __restrict__ b_hid, const float* __restrict__ b_ih0,
    const float* __restrict__ b_hh0, float* __restrict__ Wc,
    float* __restrict__ bc) {
  const int r = blockIdx.x;   // 0..1023
  const int f = threadIdx.x;  // 0..63
  float s = 0.f;
  for (int k = 0; k < H_; ++k) s += W_ih0[r * H_ + k] * W_hid[k * F_ + f];
  Wc[r * F_ + f] = s;
  if (f == 0) {
    float sb = 0.f;
    for (int k = 0; k < H_; ++k) sb += W_ih0[r * H_ + k] * b_hid[k];
    bc[r] = sb + b_ih0[r] + b_hh0[r];
  }
}

__global__ void __launch_bounds__(256) zero_kernel(float* __restrict__ p, long n) {
  long i = (long)blockIdx.x * 256 + threadIdx.x;
  if (i < n) p[i] = 0.f;
}

// ---------------------------------------------------------------------------
// C[M,N] = A[M,K] @ W[N,K]^T + bias0 + bias1  (row-major fp32)
// Block = 8 waves. All 8 waves share one 32-row A tile, staged in LDS via
// async global->LDS. Each wave computes 32(M) x 64(N): 8 accumulators.
// Requires: M%32==0, N%512==0 (nStrips multiple of 8), K%4==0, K<=256.
// ---------------------------------------------------------------------------
__global__ void __launch_bounds__(256) wmma_gemm_nt(
    const float* __restrict__ A, const float* __restrict__ W,
    const float* __restrict__ bias0, const float* __restrict__ bias1,
    float* __restrict__ C, int M, int N, int K) {
  __shared__ float As[32 * 256];  // 32 KB max (K<=256)

  const int lane = threadIdx.x & 31;
  const int wave = threadIdx.x >> 5;
  const int lm = lane & 15;
  const int half = lane >> 4;
  const int nStrips = N >> 6;
  const int blocksPerM = nStrips >> 3;  // 8 strips per block
  const long mTile = blockIdx.x / blocksPerM;
  const int strip = (blockIdx.x % blocksPerM) * 8 + wave;
  const int n0 = strip * 64;
  const long row0 = mTile * 32;

  // ---- stage contiguous 32xK A tile into LDS (async if available) ----
  {
    const float* gsrc = A + row0 * K;
    const int total4 = (32 * K) >> 2;  // float4 count
    for (int i = threadIdx.x; i < total4; i += 256)
      stage16(gsrc + i * 4, As + i * 4);
    async_wait_all();
  }
  __syncthreads();

  v8f acc[2][4];
#pragma unroll
  for (int mi = 0; mi < 2; ++mi)
#pragma unroll
    for (int j = 0; j < 4; ++j) acc[mi][j] = v8zero();

  const float* la0 = As + lm * K + 2 * half;
  const float* la1 = As + (16 + lm) * K + 2 * half;
  const float* w0 = W + (long)(n0 + 0 + lm) * K + 2 * half;
  const float* w1 = W + (long)(n0 + 16 + lm) * K + 2 * half;
  const float* w2 = W + (long)(n0 + 32 + lm) * K + 2 * half;
  const float* w3 = W + (long)(n0 + 48 + lm) * K + 2 * half;

  // software pipeline: B frags for step k+4 load while step-k WMMAs issue.
  // Final iteration peeled -> prefetch offset is an affine induction, so the
  // compiler can use immediate global_load offsets (no per-load 64-bit adds).
  v2f b0 = *(const v2f*)(w0);
  v2f b1 = *(const v2f*)(w1);
  v2f b2 = *(const v2f*)(w2);
  v2f b3 = *(const v2f*)(w3);

  int k0 = 0;
  for (; k0 < K - 4; k0 += 4) {
    const int kn = k0 + 4;
    v2f nb0 = *(const v2f*)(w0 + kn);
    v2f nb1 = *(const v2f*)(w1 + kn);
    v2f nb2 = *(const v2f*)(w2 + kn);
    v2f nb3 = *(const v2f*)(w3 + kn);
    v2f a0 = *(const v2f*)(la0 + k0);  // ds_load_b64
    v2f a1 = *(const v2f*)(la1 + k0);
    acc[0][0] = wmma4(a0, b0, acc[0][0]);
    acc[0][1] = wmma4(a0, b1, acc[0][1]);
    acc[0][2] = wmma4(a0, b2, acc[0][2]);
    acc[0][3] = wmma4(a0, b3, acc[0][3]);
    acc[1][0] = wmma4(a1, b0, acc[1][0]);
    acc[1][1] = wmma4(a1, b1, acc[1][1]);
    acc[1][2] = wmma4(a1, b2, acc[1][2]);
    acc[1][3] = wmma4(a1, b3, acc[1][3]);
    b0 = nb0; b1 = nb1; b2 = nb2; b3 = nb3;
  }
  {  // peeled last k-step
    v2f a0 = *(const v2f*)(la0 + k0);
    v2f a1 = *(const v2f*)(la1 + k0);
    acc[0][0] = wmma4(a0, b0, acc[0][0]);
    acc[0][1] = wmma4(a0, b1, acc[0][1]);
    acc[0][2] = wmma4(a0, b2, acc[0][2]);
    acc[0][3] = wmma4(a0, b3, acc[0][3]);
    acc[1][0] = wmma4(a1, b0, acc[1][0]);
    acc[1][1] = wmma4(a1, b1, acc[1][1]);
    acc[1][2] = wmma4(a1, b2, acc[1][2]);
    acc[1][3] = wmma4(a1, b3, acc[1][3]);
  }

#pragma unroll
  for (int mi = 0; mi < 2; ++mi) {
#pragma unroll
    for (int j = 0; j < 4; ++j) {
      const int n = n0 + j * 16 + lm;
      const float bs = (bias0 ? bias0[n] : 0.f) + (bias1 ? bias1[n] : 0.f);
#pragma unroll
      for (int e = 0; e < 8; ++e) {
        const long r = row0 + mi * 16 + e + 8 * half;
        C[r * N + n] = acc[mi][j][e] + bs;
      }
    }
  }
}

// ---------------------------------------------------------------------------
// One LSTM timestep: gates = gx[:,t,:] + h_{t-1} @ Whh^T ; cell update.
// Wave tile: 32(batch) x 16(hidden), 4 gates -> 8 accumulators.
// 64 waves = 8 blocks x 8 waves; the 8 waves of a block share one 32-row
// h_prev tile staged in LDS (async).
// ---------------------------------------------------------------------------
__global__ void __launch_bounds__(256) lstm_step(
    const float* __restrict__ gx,     // [B*T, 4H], row b*T + t
    const float* __restrict__ hprev,  // element [b][k] = hprev[b*stride + k]
    long hprevStride,
    const float* __restrict__ Whh,    // [4H, H]
    float* __restrict__ hseq,         // [B, T, H]
    float* __restrict__ cstate,       // [B, H]
    int t) {
  __shared__ float Hs[32 * 256];  // 32 KB

  const int lane = threadIdx.x & 31;
  const int wave = threadIdx.x >> 5;
  const int lm = lane & 15;
  const int half = lane >> 4;
  const int wg = blockIdx.x * 8 + wave;  // 0..63
  const int bTile = wg >> 4;             // 0..3  (shared by whole block)
  const int n0 = (wg & 15) * 16;         // hidden tile origin
  const int row0 = bTile * 32;

  // ---- stage h_prev rows row0..row0+31 (strided) into LDS ----
  {
    for (int i = threadIdx.x; i < 32 * 64; i += 256) {  // 64 float4 per row
      const int row = i >> 6;
      const int c4 = i & 63;
      stage16(hprev + (long)(row0 + row) * hprevStride + c4 * 4,
              Hs + row * 256 + c4 * 4);
    }
    async_wait_all();
  }
  __syncthreads();

  v8f acc[2][4];
#pragma unroll
  for (int mi = 0; mi < 2; ++mi)
#pragma unroll
    for (int q = 0; q < 4; ++q) acc[mi][q] = v8zero();

  const float* la0 = Hs + lm * 256 + 2 * half;
  const float* la1 = Hs + (16 + lm) * 256 + 2 * half;
  const float* wq0 = Whh + (long)(0 * H_ + n0 + lm) * H_ + 2 * half;
  const float* wq1 = Whh + (long)(1 * H_ + n0 + lm) * H_ + 2 * half;
  const float* wq2 = Whh + (long)(2 * H_ + n0 + lm) * H_ + 2 * half;
  const float* wq3 = Whh + (long)(3 * H_ + n0 + lm) * H_ + 2 * half;

  v2f b0 = *(const v2f*)(wq0);
  v2f b1 = *(const v2f*)(wq1);
  v2f b2 = *(const v2f*)(wq2);
  v2f b3 = *(const v2f*)(wq3);

  int k0 = 0;
  for (; k0 < H_ - 4; k0 += 4) {
    const int kn = k0 + 4;
    v2f nb0 = *(const v2f*)(wq0 + kn);
    v2f nb1 = *(const v2f*)(wq1 + kn);
    v2f nb2 = *(const v2f*)(wq2 + kn);
    v2f nb3 = *(const v2f*)(wq3 + kn);
    v2f a0 = *(const v2f*)(la0 + k0);
    v2f a1 = *(const v2f*)(la1 + k0);
    acc[0][0] = wmma4(a0, b0, acc[0][0]);
    acc[0][1] = wmma4(a0, b1, acc[0][1]);
    acc[0][2] = wmma4(a0, b2, acc[0][2]);
    acc[0][3] = wmma4(a0, b3, acc[0][3]);
    acc[1][0] = wmma4(a1, b0, acc[1][0]);
    acc[1][1] = wmma4(a1, b1, acc[1][1]);
    acc[1][2] = wmma4(a1, b2, acc[1][2]);
    acc[1][3] = wmma4(a1, b3, acc[1][3]);
    b0 = nb0; b1 = nb1; b2 = nb2; b3 = nb3;
  }
  {  // peeled last k-step
    v2f a0 = *(const v2f*)(la0 + k0);
    v2f a1 = *(const v2f*)(la1 + k0);
    acc[0][0] = wmma4(a0, b0, acc[0][0]);
    acc[0][1] = wmma4(a0, b1, acc[0][1]);
    acc[0][2] = wmma4(a0, b2, acc[0][2]);
    acc[0][3] = wmma4(a0, b3, acc[0][3]);
    acc[1][0] = wmma4(a1, b0, acc[1][0]);
    acc[1][1] = wmma4(a1, b1, acc[1][1]);
    acc[1][2] = wmma4(a1, b2, acc[1][2]);
    acc[1][3] = wmma4(a1, b3, acc[1][3]);
  }

#pragma unroll
  for (int mi = 0; mi < 2; ++mi) {
#pragma unroll
    for (int e = 0; e < 8; ++e) {
      const int b = row0 + mi * 16 + e + 8 * half;
      const int n = n0 + lm;
      const float* grow = gx + ((long)b * T_ + t) * G4H_;
      const float iv = sigf(acc[mi][0][e] + grow[0 * H_ + n]);
      const float fv = sigf(acc[mi][1][e] + grow[1 * H_ + n]);
      const float gv = tanhf(acc[mi][2][e] + grow[2 * H_ + n]);
      const float ov = sigf(acc[mi][3][e] + grow[3 * H_ + n]);
      const long ci = (long)b * H_ + n;
      const float cn = fv * cstate[ci] + iv * gv;
      cstate[ci] = cn;
      hseq[((long)b * T_ + t) * H_ + n] = ov * tanhf(cn);
    }
  }
}

// ---------------------------------------------------------------------------
// y[b,tp,o] = relu(h2[b, T-1-PRED+tp, :]) @ W_out[F,H]^T + b_out
// M = B*PRED = 12288 rows, N = 64, K = 256. 768 waves = 96 blocks.
// ---------------------------------------------------------------------------
__global__ void __launch_bounds__(256) out_gemm(
    const float* __restrict__ h2seq, const float* __restrict__ Wout,
    const float* __restrict__ bout, float* __restrict__ y) {
  const int lane = threadIdx.x & 31;
  const int wave = threadIdx.x >> 5;
  const int lm = lane & 15;
  const int half = lane >> 4;
  const long wg = (long)blockIdx.x * 8 + wave;
  const long row0 = wg * 16;

  const long r = row0 + lm;
  const int bIdx = (int)(r / PRED_);
  const int tp = (int)(r % PRED_);
  const float* ap =
      h2seq + ((long)bIdx * T_ + (T_ - 1 - PRED_) + tp) * H_ + 2 * half;
  const float* w0 = Wout + (long)(0 + lm) * H_ + 2 * half;
  const float* w1 = Wout + (long)(16 + lm) * H_ + 2 * half;
  const float* w2 = Wout + (long)(32 + lm) * H_ + 2 * half;
  const float* w3 = Wout + (long)(48 + lm) * H_ + 2 * half;

  v8f acc0 = v8zero(), acc1 = v8zero(), acc2 = v8zero(), acc3 = v8zero();

  v2f a, b0, b1, b2, b3;
  a.x = fmaxf(ap[0], 0.f);
  a.y = fmaxf(ap[1], 0.f);
  b0 = *(const v2f*)(w0);
  b1 = *(const v2f*)(w1);
  b2 = *(const v2f*)(w2);
  b3 = *(const v2f*)(w3);

  int k0 = 0;
  for (; k0 < H_ - 4; k0 += 4) {
    const int kn = k0 + 4;
    v2f na, nb0, nb1, nb2, nb3;
    na.x = fmaxf(ap[kn + 0], 0.f);  // fused ReLU on Hbuf
    na.y = fmaxf(ap[kn + 1], 0.f);
    nb0 = *(const v2f*)(w0 + kn);
    nb1 = *(const v2f*)(w1 + kn);
    nb2 = *(const v2f*)(w2 + kn);
    nb3 = *(const v2f*)(w3 + kn);
    acc0 = wmma4(a, b0, acc0);
    acc1 = wmma4(a, b1, acc1);
    acc2 = wmma4(a, b2, acc2);
    acc3 = wmma4(a, b3, acc3);
    a = na; b0 = nb0; b1 = nb1; b2 = nb2; b3 = nb3;
  }
  {  // peeled last k-step
    acc0 = wmma4(a, b0, acc0);
    acc1 = wmma4(a, b1, acc1);
    acc2 = wmma4(a, b2, acc2);
    acc3 = wmma4(a, b3, acc3);
  }

#pragma unroll
  for (int j = 0; j < 4; ++j) {
    v8f acc = (j == 0) ? acc0 : (j == 1) ? acc1 : (j == 2) ? acc2 : acc3;
    const int n = j * 16 + lm;
    const float bs = bout[n];
#pragma unroll
    for (int e = 0; e < 8; ++e) {
      const long ro = row0 + e + 8 * half;  // = b*PRED + tp
      y[ro * F_ + n] = acc[e] + bs;
    }
  }
}

// ---------------------------------------------------------------------------
extern "C" void kernel_launch(void* const* d_in, const int* in_sizes, int n_in,
                              void* d_out, int out_size, void* d_ws,
                              size_t ws_size, hipStream_t stream) {
  const float* x     = (const float*)d_in[0];
  const float* W_hid = (const float*)d_in[1];
  const float* b_hid = (const float*)d_in[2];
  const float* W_ih0 = (const float*)d_in[3];
  const float* W_hh0 = (const float*)d_in[4];
  const float* b_ih0 = (const float*)d_in[5];
  const float* b_hh0 = (const float*)d_in[6];
  const float* W_ih1 = (const float*)d_in[7];
  const float* W_hh1 = (const float*)d_in[8];
  const float* b_ih1 = (const float*)d_in[9];
  const float* b_hh1 = (const float*)d_in[10];
  const float* W_out = (const float*)d_in[11];
  const float* b_out = (const float*)d_in[12];
  float* out = (float*)d_out;

  char* ws = (char*)d_ws;
  size_t off = 0;
  auto alloc = [&](size_t bytes) -> float* {
    float* p = (float*)(ws + off);
    off = (off + bytes + 255) & ~(size_t)255;
    return p;
  };
  float* gx = alloc((size_t)B_ * T_ * G4H_ * 4);  // 512 MB, reused by both layers
  float* h1 = alloc((size_t)B_ * T_ * H_ * 4);    // 128 MB
  float* h2 = alloc((size_t)B_ * T_ * H_ * 4);    // 128 MB
  float* c0 = alloc((size_t)B_ * H_ * 4);
  float* c1 = alloc((size_t)B_ * H_ * 4);
  float* hz = alloc((size_t)B_ * H_ * 4);         // zero h_{-1}
  float* Wc = alloc((size_t)G4H_ * F_ * 4);
  float* bc = alloc((size_t)G4H_ * 4);

  {  // zero c0, c1, hz (contiguous)
    long n = 3L * B_ * H_;
    zero_kernel<<<dim3((unsigned)((n + 255) / 256)), dim3(256), 0, stream>>>(c0, n);
  }
  prep_kernel<<<dim3(G4H_), dim3(64), 0, stream>>>(W_ih0, W_hid, b_hid, b_ih0,
                                                   b_hh0, Wc, bc);

  const int MBT = B_ * T_;  // 131072
  // gx (layer 0) = x @ Wc^T + bc   [131072 x 1024, K=64]
  {
    unsigned blocks = (unsigned)(((long)(MBT / 32) * (G4H_ / 64)) / 8);  // 8192
    wmma_gemm_nt<<<dim3(blocks), dim3(256), 0, stream>>>(
        x, Wc, bc, (const float*)nullptr, gx, MBT, G4H_, F_);
  }
  // layer-0 recurrent scan
  for (int t = 0; t < T_; ++t) {
    const float* hp = (t == 0) ? hz : (h1 + (long)(t - 1) * H_);
    long stride = (t == 0) ? (long)H_ : (long)T_ * H_;
    lstm_step<<<dim3(8), dim3(256), 0, stream>>>(gx, hp, stride, W_hh0, h1, c0, t);
  }
  // gx (layer 1) = h1 @ W_ih1^T + (b_ih1 + b_hh1)   [131072 x 1024, K=256]
  {
    unsigned blocks = (unsigned)(((long)(MBT / 32) * (G4H_ / 64)) / 8);
    wmma_gemm_nt<<<dim3(blocks), dim3(256), 0, stream>>>(
        h1, W_ih1, b_ih1, b_hh1, gx, MBT, G4H_, H_);
  }
  // layer-1 scan: only t = 0..T-2 are ever consumed (h2[:, :-1])
  for (int t = 0; t < T_ - 1; ++t) {
    const float* hp = (t == 0) ? hz : (h2 + (long)(t - 1) * H_);
    long stride = (t == 0) ? (long)H_ : (long)T_ * H_;
    lstm_step<<<dim3(8), dim3(256), 0, stream>>>(gx, hp, stride, W_hh1, h2, c1, t);
  }
  // y = relu(h2[:, T-1-PRED : T-1, :]) @ W_out^T + b_out
  out_gemm<<<dim3(96), dim3(256), 0, stream>>>(h2, W_out, b_out, out);
}